// GridSamplePScan_90288802497258
// MI455X (gfx1250) — compile-verified
//
#include <hip/hip_runtime.h>
#include <hip/hip_bf16.h>
#include <math.h>

// Problem constants (fixed by reference's setup_inputs)
#define BB 4
#define LL 32
#define CC 16
#define HH 128
#define WW 128
#define HWPX (HH * WW)

// ---------------------------------------------------------------------------
// CDNA5 async global->LDS copy helpers (gfx1250).
// The builtin's param 1 is a global-AS pointer to a 4-int vector (per clang
// diagnostic: "int vector __device__ *"), param 2 the LDS-AS counterpart.
// ---------------------------------------------------------------------------
typedef int v4i __attribute__((ext_vector_type(4)));
typedef __attribute__((address_space(1))) float as1_float;
typedef __attribute__((address_space(3))) float as3_float;
typedef __attribute__((address_space(1))) v4i   as1_v4i;
typedef __attribute__((address_space(3))) v4i   as3_v4i;
typedef __attribute__((address_space(3))) char  as3_char;

__device__ __forceinline__ void async_copy_b128(const float* gsrc, float* lds_dst) {
#if defined(__gfx1250__)
#if defined(__has_builtin) && __has_builtin(__builtin_amdgcn_global_load_async_to_lds_b128)
    as1_float* gp = (as1_float*)const_cast<float*>(gsrc); // generic -> AS1
    as3_float* lp = (as3_float*)lds_dst;                  // generic -> AS3
    __builtin_amdgcn_global_load_async_to_lds_b128(
        (as1_v4i*)gp, (as3_v4i*)lp, 0, 0);
#else
    unsigned lds_addr = (unsigned)(unsigned long long)(as3_char*)(void*)lds_dst;
    unsigned long long gaddr = (unsigned long long)gsrc;
    asm volatile("global_load_async_to_lds_b128 %0, %1, off"
                 :: "v"(lds_addr), "v"(gaddr)
                 : "memory");
#endif
#else
    // Host-pass / non-gfx1250 placeholder (never executed on device target)
    (void)gsrc; (void)lds_dst;
#endif
}

__device__ __forceinline__ void wait_async0() {
#if defined(__gfx1250__)
#if defined(__has_builtin) && __has_builtin(__builtin_amdgcn_s_wait_asynccnt)
    __builtin_amdgcn_s_wait_asynccnt(0);
#else
    asm volatile("s_wait_asynccnt 0" ::: "memory");
#endif
#endif
}

// ---------------------------------------------------------------------------
// Grid composition (border-padded bilinear + mask). Layout: (B,L,H,W,2) f32.
// dst[:, :step] = src[:, :step]; dst[:, l>=step] = compose(src[l-step], src[l]).
// Grid planes are tiny (128 KB/slice) and L2-resident -> direct global gathers.
// ---------------------------------------------------------------------------
__global__ void grid_compose_kernel(const float* __restrict__ g_src,
                                    float* __restrict__ g_dst,
                                    int step) {
    int idx = blockIdx.x * blockDim.x + threadIdx.x;   // over B*L*H*W
    if (idx >= BB * LL * HWPX) return;
    int p   = idx / HWPX;          // b*L + l
    int pix = idx - p * HWPX;
    int l   = p & (LL - 1);

    const float2* src = (const float2*)g_src;
    float2*       dst = (float2*)g_dst;
    float2 g = src[(size_t)p * HWPX + pix];

    if (l < step) {
        dst[(size_t)p * HWPX + pix] = g;
        return;
    }

    const float2* prev = src + (size_t)(p - step) * HWPX;

    float mask = (fabsf(g.x) <= 1.0f && fabsf(g.y) <= 1.0f) ? 1.0f : 0.0f;

    // border padding: clip unnormalized coords before corner computation
    float ix = (g.x + 1.0f) * 0.5f * (float)(WW - 1);
    float iy = (g.y + 1.0f) * 0.5f * (float)(HH - 1);
    ix = fminf(fmaxf(ix, 0.0f), (float)(WW - 1));
    iy = fminf(fmaxf(iy, 0.0f), (float)(HH - 1));

    float x0f = floorf(ix), y0f = floorf(iy);
    float wx1 = ix - x0f, wy1 = iy - y0f;
    float wx0 = 1.0f - wx1, wy0 = 1.0f - wy1;

    int x0 = (int)x0f, y0 = (int)y0f;
    int xc0 = min(max(x0, 0), WW - 1);
    int xc1 = min(max(x0 + 1, 0), WW - 1);
    int yc0 = min(max(y0, 0), HH - 1);
    int yc1 = min(max(y0 + 1, 0), HH - 1);

    float2 c00 = prev[yc0 * WW + xc0];
    float2 c01 = prev[yc0 * WW + xc1];
    float2 c10 = prev[yc1 * WW + xc0];
    float2 c11 = prev[yc1 * WW + xc1];

    float ox = (c00.x * wx0 + c01.x * wx1) * wy0 + (c10.x * wx0 + c11.x * wx1) * wy1;
    float oy = (c00.y * wx0 + c01.y * wx1) * wy0 + (c10.y * wx0 + c11.y * wx1) * wy1;

    float2 o;
    o.x = mask * ox + (1.0f - mask) * (-2.0f);
    o.y = mask * oy + (1.0f - mask) * (-2.0f);
    dst[(size_t)p * HWPX + pix] = o;
}

// ---------------------------------------------------------------------------
// Image warp + add. One 256-thread block per (b,l,c) channel plane.
// Stages the 64 KB img_prev plane into LDS via async b128 copies (ASYNCcnt),
// then serves the 4 bilinear gathers per pixel from LDS.
// ---------------------------------------------------------------------------
__global__ void __launch_bounds__(256)
warp_add_kernel(const float* __restrict__ img_src,
                const float* __restrict__ grid_src,
                float* __restrict__ img_dst,
                int step) {
    extern __shared__ float plane[];          // HWPX floats = 64 KB (dynamic)
    const int c = blockIdx.y;                 // channel
    const int p = blockIdx.z;                 // b*L + l
    const int l = p & (LL - 1);
    const int t = threadIdx.x;                // 0..255

    const float* src_cur = img_src + ((size_t)p * CC + c) * HWPX;
    float*       dst     = img_dst + ((size_t)p * CC + c) * HWPX;

    if (l < step) {
        // pass-through copy of the scan's unchanged prefix
        const float4* s4 = (const float4*)src_cur;
        float4*       d4 = (float4*)dst;
#pragma unroll
        for (int j = 0; j < (HWPX / 4) / 256; ++j)
            d4[t + j * 256] = s4[t + j * 256];
        return;
    }

    const float* src_prev = img_src + ((size_t)(p - step) * CC + c) * HWPX;

    // ---- async stage prev channel plane -> LDS (4096 x b128) ----
#pragma unroll
    for (int j = 0; j < (HWPX / 4) / 256; ++j) {   // 16 per thread
        int k = t + j * 256;                        // float4 index
        async_copy_b128(src_prev + 4 * k, &plane[4 * k]);
    }
    wait_async0();
    __syncthreads();

    const float2* gcur = (const float2*)grid_src + (size_t)p * HWPX;

    for (int j = 0; j < HWPX / 256; ++j) {          // 64 pixels/thread
        int pix = t + j * 256;
        float2 g = gcur[pix];

        // zeros padding: no pre-clip; per-corner validity masks the weight
        float ix = (g.x + 1.0f) * 0.5f * (float)(WW - 1);
        float iy = (g.y + 1.0f) * 0.5f * (float)(HH - 1);
        float x0f = floorf(ix), y0f = floorf(iy);
        float wx1 = ix - x0f, wy1 = iy - y0f;
        float wx0 = 1.0f - wx1, wy0 = 1.0f - wy1;

        int x0 = (int)x0f, y0 = (int)y0f;
        bool vx0 = (x0 >= 0) && (x0 <= WW - 1);
        bool vx1 = (x0 + 1 >= 0) && (x0 + 1 <= WW - 1);
        bool vy0 = (y0 >= 0) && (y0 <= HH - 1);
        bool vy1 = (y0 + 1 >= 0) && (y0 + 1 <= HH - 1);

        int xc0 = min(max(x0, 0), WW - 1);
        int xc1 = min(max(x0 + 1, 0), WW - 1);
        int yc0 = min(max(y0, 0), HH - 1);
        int yc1 = min(max(y0 + 1, 0), HH - 1);

        float w00 = (vx0 && vy0) ? wy0 * wx0 : 0.0f;
        float w01 = (vx1 && vy0) ? wy0 * wx1 : 0.0f;
        float w10 = (vx0 && vy1) ? wy1 * wx0 : 0.0f;
        float w11 = (vx1 && vy1) ? wy1 * wx1 : 0.0f;

        float val = plane[(yc0 << 7) + xc0] * w00
                  + plane[(yc0 << 7) + xc1] * w01
                  + plane[(yc1 << 7) + xc0] * w10
                  + plane[(yc1 << 7) + xc1] * w11;

        dst[pix] = val + src_cur[pix];
    }
}

// ---------------------------------------------------------------------------
// Host launcher: 5 scan rounds, ping-pong buffers, all on `stream`.
// ws layout: [gridA 16MB][gridB 16MB][imgW 128MB]  -> 160 MB needed.
// ---------------------------------------------------------------------------
extern "C" void kernel_launch(void* const* d_in, const int* in_sizes, int n_in,
                              void* d_out, int out_size, void* d_ws, size_t ws_size,
                              hipStream_t stream) {
    (void)in_sizes; (void)n_in; (void)out_size; (void)ws_size;

    const float* g_in = (const float*)d_in[0];   // grids  (B,L,H,W,2)
    const float* i_in = (const float*)d_in[1];   // images (B,L,C,H,W)
    float* out = (float*)d_out;

    const size_t GRID_BYTES = (size_t)BB * LL * HWPX * 2 * sizeof(float);   // 16 MB
    char* ws = (char*)d_ws;
    float* gridA = (float*)ws;
    float* gridB = (float*)(ws + GRID_BYTES);
    float* imgW  = (float*)(ws + 2 * GRID_BYTES);

    const int steps[5] = {1, 2, 4, 8, 16};
    const float* gsrc[5] = {g_in, gridA, gridB, gridA, gridB};
    float*       gdst[5] = {gridA, gridB, gridA, gridB, nullptr}; // last grid unused
    const float* isrc[5] = {i_in, out, imgW, out, imgW};
    float*       idst[5] = {out, imgW, out, imgW, out};

    const int npx = BB * LL * HWPX;
    const dim3 gridGC((npx + 255) / 256);
    const dim3 gridWA(1, CC, BB * LL);
    const size_t lds_bytes = (size_t)HWPX * sizeof(float); // 64 KB dynamic LDS

    for (int s = 0; s < 5; ++s) {
        if (gdst[s])
            grid_compose_kernel<<<gridGC, 256, 0, stream>>>(gsrc[s], gdst[s], steps[s]);
        warp_add_kernel<<<gridWA, 256, lds_bytes, stream>>>(isrc[s], gsrc[s], idst[s], steps[s]);
    }
}